// MatchingHead_55688545960679
// MI455X (gfx1250) — compile-verified
//
#include <hip/hip_runtime.h>
#include <math.h>

// ---------------- problem constants ----------------
namespace {
constexpr int kB  = 256;
constexpr int kN  = 64;
constexpr int kD  = 256;
constexpr int kH  = 8;
constexpr int kDH = 32;
constexpr int kL  = 18;
constexpr int kIters = 200;
constexpr int kM  = kB * kN;      // 16384 token rows
constexpr int kZd = kN + 1;       // 65
constexpr int kKC = 32;           // K chunk staged per double-buffer step
constexpr int kPad = 36;          // LDS row pitch (conflict-free, 16B-aligned)
}

typedef float v2f __attribute__((ext_vector_type(2)));
typedef float v8f __attribute__((ext_vector_type(8)));

// =====================================================================
// GEMM via V_WMMA_F32_16X16X4_F32, LDS-staged + double-buffered.
//   C[kM x N] = act( A1 @ W[0:K,:] (+ A2 @ W[K:2K,:]) + bias )
// Block: 256 thr (8 waves) -> 128(M) x 64(N) tile. Per 32-K chunk:
//   As[128][36]  row-major A tile      (ds_load_b64 A fragment)
//   Ws[64][36]   TRANSPOSED W tile     (ds_load_b64 B fragment: rows k,k+1)
// Fragment layouts per CDNA5 ISA 7.12.2:
//   A 16x4 : lane<16 -> M=lane, K={k0,k0+1} ; lane>=16 -> K={k0+2,k0+3}
//   B 4x16 : VGPR j, lane l -> K = k0 + 2*(l/16) + j, N = l%16
//   C 16x16: VGPR j, lane l -> M = j + 8*(l/16),      N = l%16
// =====================================================================
__global__ __launch_bounds__(256) void mh_gemm_wmma_f32(
    const float* __restrict__ A1, const float* __restrict__ A2,
    const float* __restrict__ W,  const float* __restrict__ bias,
    float* __restrict__ C, int K, int N, int do_relu)
{
  __shared__ float As[2][128 * kPad];   // 36.9 KB
  __shared__ float Ws[2][64 * kPad];    // 18.4 KB

  const int tid  = threadIdx.x;
  const int lane = tid & 31;
  const int wv   = tid >> 5;
  const int row0 = blockIdx.x * 128;    // block M origin
  const int nb   = blockIdx.y * 64;     // block N origin
  const int koff = (lane >> 4) << 1;    // 0 or 2
  const int l16  = lane & 15;

  const v8f vzero = {0.f,0.f,0.f,0.f,0.f,0.f,0.f,0.f};
  v8f acc[4];
#pragma unroll
  for (int t = 0; t < 4; ++t) acc[t] = vzero;

  const int cpk     = K / kKC;                       // chunks per pass
  const int nchunks = (A2 ? 2 : 1) * cpk;

  float4 areg[4];
  float4 wreg[2];

  auto gload = [&](int ch) {
    const int pass = ch / cpk;
    const int kc   = (ch % cpk) * kKC;
    const float* Ap = (pass == 0) ? A1 : A2;
    const float* Wp = W + (size_t)pass * K * N;
#pragma unroll
    for (int i = 0; i < 4; ++i) {                    // A: 128x32 = 1024 float4
      const int f = tid + i * 256;
      const int r = f >> 3, c4 = f & 7;
      areg[i] = *(const float4*)(Ap + (size_t)(row0 + r) * K + kc + c4 * 4);
    }
#pragma unroll
    for (int i = 0; i < 2; ++i) {                    // W: 32x64 = 512 float4
      const int f = tid + i * 256;
      const int r = f >> 4, c4 = f & 15;
      wreg[i] = *(const float4*)(Wp + (size_t)(kc + r) * N + nb + c4 * 4);
    }
  };

  auto stage = [&](int buf) {
#pragma unroll
    for (int i = 0; i < 4; ++i) {
      const int f = tid + i * 256;
      const int r = f >> 3, c4 = f & 7;
      *(float4*)&As[buf][r * kPad + c4 * 4] = areg[i];   // 16B aligned (36*4=144)
    }
#pragma unroll
    for (int i = 0; i < 2; ++i) {                    // transpose W into LDS
      const int f = tid + i * 256;
      const int r = f >> 4, c = (f & 15) * 4;
      Ws[buf][(c + 0) * kPad + r] = wreg[i].x;
      Ws[buf][(c + 1) * kPad + r] = wreg[i].y;
      Ws[buf][(c + 2) * kPad + r] = wreg[i].z;
      Ws[buf][(c + 3) * kPad + r] = wreg[i].w;
    }
  };

  gload(0);
  stage(0);
  __syncthreads();

  for (int ch = 0; ch < nchunks; ++ch) {
    const int buf = ch & 1;
    if (ch + 1 < nchunks) gload(ch + 1);             // overlap with compute

    const float* Ab = &As[buf][(wv * 16 + l16) * kPad + koff];
#pragma unroll
    for (int k0 = 0; k0 < kKC; k0 += 4) {
      const v2f a = *(const v2f*)(Ab + k0);
#pragma unroll
      for (int t = 0; t < 4; ++t) {
        const v2f bv = *(const v2f*)&Ws[buf][(t * 16 + l16) * kPad + k0 + koff];
        acc[t] = __builtin_amdgcn_wmma_f32_16x16x4_f32(
            false, a, false, bv, (short)0, acc[t], false, false);
      }
    }

    if (ch + 1 < nchunks) {
      stage(buf ^ 1);                                // other buffer: no barrier needed first
      __syncthreads();
    }
  }

#pragma unroll
  for (int t = 0; t < 4; ++t) {
#pragma unroll
    for (int j = 0; j < 8; ++j) {
      const int m = row0 + wv * 16 + j + 8 * (lane >> 4);
      const int n = nb + t * 16 + l16;
      float val = acc[t][j];
      if (bias) val += bias[n];
      if (do_relu) val = fmaxf(val, 0.f);
      C[(size_t)m * N + n] = val;
    }
  }
}

// =====================================================================
// Linear attention for one (batch, head): tiny 64x32 problem in LDS.
// =====================================================================
__global__ __launch_bounds__(256) void mh_linattn(
    const float* __restrict__ Qg, const float* __restrict__ Kg,
    const float* __restrict__ Vg, float* __restrict__ Og)
{
  __shared__ float Qs[kN * 33], Ks[kN * 33], Vs[kN * 33];
  __shared__ float KV[kDH * 33];
  __shared__ float Ksum[kDH];
  __shared__ float Zden[kN];
  const int b = blockIdx.x / kH;
  const int h = blockIdx.x % kH;
  const int tid = threadIdx.x;
  const size_t base = (size_t)b * kN * kD + h * kDH;

  for (int idx = tid; idx < kN * kDH; idx += 256) {
    const int n = idx >> 5, d = idx & 31;
    const size_t g = base + (size_t)n * kD + d;
    const float q = Qg[g], k = Kg[g];
    Qs[n * 33 + d] = q > 0.f ? q + 1.f : __expf(q);   // elu(x)+1
    Ks[n * 33 + d] = k > 0.f ? k + 1.f : __expf(k);
    Vs[n * 33 + d] = Vg[g] * (1.0f / kN);             // v / v_len
  }
  __syncthreads();
  for (int idx = tid; idx < kDH * kDH; idx += 256) {
    const int d = idx >> 5, e = idx & 31;
    float s = 0.f;
    for (int m = 0; m < kN; ++m) s += Ks[m * 33 + d] * Vs[m * 33 + e];
    KV[d * 33 + e] = s;
  }
  if (tid < kDH) {
    float s = 0.f;
    for (int m = 0; m < kN; ++m) s += Ks[m * 33 + tid];
    Ksum[tid] = s;
  }
  __syncthreads();
  if (tid < kN) {
    float s = 0.f;
    for (int d = 0; d < kDH; ++d) s += Qs[tid * 33 + d] * Ksum[d];
    Zden[tid] = 1.f / (s + 1e-6f);
  }
  __syncthreads();
  for (int idx = tid; idx < kN * kDH; idx += 256) {
    const int n = idx >> 5, e = idx & 31;
    float s = 0.f;
    for (int d = 0; d < kDH; ++d) s += Qs[n * 33 + d] * KV[d * 33 + e];
    Og[base + (size_t)n * kD + e] = s * Zden[n] * (float)kN;
  }
}

// =====================================================================
// LayerNorm over D=256, one wave32 per row (8 elems/lane), shuffle reduce.
// =====================================================================
__global__ __launch_bounds__(256) void mh_layernorm(
    const float* __restrict__ src, float* __restrict__ dst,
    const float* __restrict__ g, const float* __restrict__ bta)
{
  const int row  = blockIdx.x * 8 + (threadIdx.x >> 5);
  const int lane = threadIdx.x & 31;
  const float* p = src + (size_t)row * kD;
  float x[8];
  float s = 0.f;
#pragma unroll
  for (int j = 0; j < 8; ++j) { x[j] = p[lane + j * 32]; s += x[j]; }
#pragma unroll
  for (int off = 16; off > 0; off >>= 1) s += __shfl_xor(s, off, 32);
  const float mean = s * (1.0f / kD);
  float v = 0.f;
#pragma unroll
  for (int j = 0; j < 8; ++j) { const float d = x[j] - mean; v += d * d; }
#pragma unroll
  for (int off = 16; off > 0; off >>= 1) v += __shfl_xor(v, off, 32);
  const float inv = rsqrtf(v * (1.0f / kD) + 1e-5f);
#pragma unroll
  for (int j = 0; j < 8; ++j) {
    const int idx = lane + j * 32;
    dst[(size_t)row * kD + idx] = (x[j] - mean) * inv * g[idx] + bta[idx];
  }
}

__global__ void mh_add_inplace(float* __restrict__ x, const float* __restrict__ d, int n)
{
  const int i = blockIdx.x * 256 + threadIdx.x;
  if (i < n) x[i] += d[i];
}

// =====================================================================
// Geometry: warp params1 by quat/tran, build normal/offset distance maps.
// =====================================================================
__global__ __launch_bounds__(64) void mh_geom(
    const float* __restrict__ cam, const float* __restrict__ p1,
    const float* __restrict__ p2, float* __restrict__ nd, float* __restrict__ od)
{
  __shared__ float n1r[kN][3], n1rt[kN][3], o1[kN], n2s[kN][3], o2[kN];
  const int b = blockIdx.x, n = threadIdx.x;
  const float* c = cam + (size_t)b * 7;
  const float tx = c[0], ty = c[1], tz = c[2];
  float qw = c[3], qx = c[4], qy = c[5], qz = c[6];
  const float qn = rsqrtf(qw * qw + qx * qx + qy * qy + qz * qz);
  qw *= qn; qx *= qn; qy *= qn; qz *= qn;
  const float R00 = 1.f - 2.f * (qy * qy + qz * qz), R01 = 2.f * (qx * qy - qw * qz), R02 = 2.f * (qx * qz + qw * qy);
  const float R10 = 2.f * (qx * qy + qw * qz), R11 = 1.f - 2.f * (qx * qx + qz * qz), R12 = 2.f * (qy * qz - qw * qx);
  const float R20 = 2.f * (qx * qz - qw * qy), R21 = 2.f * (qy * qz + qw * qx), R22 = 1.f - 2.f * (qx * qx + qy * qy);

  {
    const float* P2 = p2 + ((size_t)b * kN + n) * 3;
    const float wx = P2[0], wy = -P2[1], wz = -P2[2];          // FLIP = [1,-1,-1]
    const float nr = sqrtf(wx * wx + wy * wy + wz * wz);
    o2[n] = nr;
    const float iv = 1.f / fmaxf(nr, 1e-12f);
    n2s[n][0] = wx * iv; n2s[n][1] = wy * iv; n2s[n][2] = wz * iv;
  }
  {
    const float* P1 = p1 + ((size_t)b * kN + n) * 3;
    const float px = P1[0], py = -P1[1], pz = -P1[2];
    const float bx = R00 * px + R01 * py + R02 * pz;
    const float by = R10 * px + R11 * py + R12 * pz;
    const float bz = R20 * px + R21 * py + R22 * pz;
    const float bb = bx * bx + by * by + bz * bz;
    const float bl = sqrtf(bb);
    const float den = 1.f / ((bl + 1e-5f) * (bl + 1e-5f));
    {
      const float sc = bb * den;
      const float rx = sc * bx, ry = sc * by, rz = sc * bz;
      const float rl = sqrtf(rx * rx + ry * ry + rz * rz);
      const float ir = 1.f / fmaxf(rl, 1e-12f);
      n1r[n][0] = rx * ir; n1r[n][1] = ry * ir; n1r[n][2] = rz * ir;
    }
    {
      const float de = (bx + tx) * bx + (by + ty) * by + (bz + tz) * bz;
      const float sc = de * den;
      const float sx = sc * bx, sy = sc * by, sz = sc * bz;
      const float sl = sqrtf(sx * sx + sy * sy + sz * sz);
      o1[n] = sl;
      const float is = 1.f / fmaxf(sl, 1e-12f);
      n1rt[n][0] = sx * is; n1rt[n][1] = sy * is; n1rt[n][2] = sz * is;
    }
  }
  __syncthreads();
  const size_t rowb = ((size_t)b * kN + n) * kN;
  for (int m = 0; m < kN; ++m) {
    const float dn = n1r[n][0] * n2s[m][0] + n1r[n][1] * n2s[m][1] + n1r[n][2] * n2s[m][2];
    nd[rowb + m] = acosf(fminf(fmaxf(dn, -1.f), 1.f)) * 57.29577951308232f;
    const float dr = n1rt[n][0] * n2s[m][0] + n1rt[n][1] * n2s[m][1] + n1rt[n][2] * n2s[m][2];
    const float o = (dr < 0.f) ? fabsf(o1[n] + o2[m]) : fabsf(o1[n] - o2[m]);
    od[rowb + m] = fminf(fmaxf(o, 1e-10f), 5.f);
  }
}

// =====================================================================
// Z assembly: scores = d1.d2/16 - od/0.5 - nd/10 ; borders = bin_score.
// =====================================================================
__global__ void mh_scores(
    const float* __restrict__ d1, const float* __restrict__ d2,
    const float* __restrict__ nd, const float* __restrict__ od,
    const float* __restrict__ bin, float* __restrict__ Z)
{
  const int idx = blockIdx.x * 256 + threadIdx.x;
  if (idx >= kB * kZd * kZd) return;
  const int b = idx / (kZd * kZd);
  const int r = idx % (kZd * kZd);
  const int i = r / kZd, j = r % kZd;
  float val;
  if (i < kN && j < kN) {
    const float4* a = (const float4*)(d1 + ((size_t)b * kN + i) * kD);
    const float4* c = (const float4*)(d2 + ((size_t)b * kN + j) * kD);
    float s = 0.f;
    for (int d = 0; d < kD / 4; ++d) {
      const float4 x = a[d], y = c[d];
      s += x.x * y.x + x.y * y.y + x.z * y.z + x.w * y.w;
    }
    const size_t g = ((size_t)b * kN + i) * kN + j;
    val = s * 0.0625f - od[g] * 2.0f - nd[g] * 0.1f;
  } else {
    val = bin[0];
  }
  Z[idx] = val;
}

// =====================================================================
// Sinkhorn: one workgroup per batch element; Z (65x65, 17KB) lives in LDS
// for all 200 iterations; wave32 shuffle logsumexp per row/column.
// =====================================================================
__global__ __launch_bounds__(256) void mh_sinkhorn(
    const float* __restrict__ Zg, float* __restrict__ out)
{
  __shared__ float Zs[kZd * 66];
  __shared__ float uu[kZd], vv[kZd];
  const int b = blockIdx.x;
  const int tid = threadIdx.x, lane = tid & 31, wv = tid >> 5;
  for (int idx = tid; idx < kZd * kZd; idx += 256)
    Zs[(idx / kZd) * 66 + (idx % kZd)] = Zg[(size_t)b * kZd * kZd + idx];
  if (tid < kZd) { uu[tid] = 0.f; vv[tid] = 0.f; }
  __syncthreads();

  const float norm    = -logf(128.f);          // -log(N1+N2)
  const float lm_last = logf(64.f) + norm;     // log(N) + norm

  for (int it = 0; it < kIters; ++it) {
    for (int i = wv; i < kZd; i += 8) {
      float a0 = Zs[i * 66 + lane] + vv[lane];
      float a1 = Zs[i * 66 + lane + 32] + vv[lane + 32];
      float a2 = (lane == 0) ? (Zs[i * 66 + 64] + vv[64]) : -INFINITY;
      float m = fmaxf(fmaxf(a0, a1), a2);
#pragma unroll
      for (int off = 16; off > 0; off >>= 1) m = fmaxf(m, __shfl_xor(m, off, 32));
      float s = __expf(a0 - m) + __expf(a1 - m) + ((lane == 0) ? __expf(a2 - m) : 0.f);
#pragma unroll
      for (int off = 16; off > 0; off >>= 1) s += __shfl_xor(s, off, 32);
      if (lane == 0) uu[i] = ((i == kN) ? lm_last : norm) - (m + __logf(s));
    }
    __syncthreads();
    for (int j = wv; j < kZd; j += 8) {
      float a0 = Zs[lane * 66 + j] + uu[lane];
      float a1 = Zs[(lane + 32) * 66 + j] + uu[lane + 32];
      float a2 = (lane == 0) ? (Zs[64 * 66 + j] + uu[64]) : -INFINITY;
      float m = fmaxf(fmaxf(a0, a1), a2);
#pragma unroll
      for (int off = 16; off > 0; off >>= 1) m = fmaxf(m, __shfl_xor(m, off, 32));
      float s = __expf(a0 - m) + __expf(a1 - m) + ((lane == 0) ? __expf(a2 - m) : 0.f);
#pragma unroll
      for (int off = 16; off > 0; off >>= 1) s += __shfl_xor(s, off, 32);
      if (lane == 0) vv[j] = ((j == kN) ? lm_last : norm) - (m + __logf(s));
    }
    __syncthreads();
  }
  for (int idx = tid; idx < kZd * kZd; idx += 256) {
    const int i = idx / kZd, j = idx % kZd;
    out[(size_t)b * kZd * kZd + idx] = Zs[i * 66 + j] + uu[i] + vv[j] - norm;
  }
}

// =====================================================================
// Host orchestration
// =====================================================================
extern "C" void kernel_launch(void* const* d_in, const int* in_sizes, int n_in,
                              void* d_out, int out_size, void* d_ws, size_t ws_size,
                              hipStream_t stream) {
  (void)in_sizes; (void)n_in; (void)out_size; (void)ws_size;
  const float* planeApp1 = (const float*)d_in[0];
  const float* planeApp2 = (const float*)d_in[1];
  const float* cam       = (const float*)d_in[2];
  const float* params1   = (const float*)d_in[3];
  const float* params2   = (const float*)d_in[4];
  const float* app_W     = (const float*)d_in[5];
  const float* app_b     = (const float*)d_in[6];
  const float* desc_W    = (const float*)d_in[7];
  const float* desc_b    = (const float*)d_in[8];
  const float* bin_score = (const float*)d_in[9];
  const float* gWq = (const float*)d_in[10];
  const float* gWk = (const float*)d_in[11];
  const float* gWv = (const float*)d_in[12];
  const float* gWm = (const float*)d_in[13];
  const float* gW1 = (const float*)d_in[14];
  const float* gW2 = (const float*)d_in[15];
  const float* g1g = (const float*)d_in[16];
  const float* g1b = (const float*)d_in[17];
  const float* g2g = (const float*)d_in[18];
  const float* g2b = (const float*)d_in[19];

  // workspace layout (floats)
  const size_t MN = (size_t)kM * kD;          // 4,194,304
  float* w  = (float*)d_ws;
  float* X0  = w;            float* X1  = X0 + MN;
  float* Qb  = X1 + MN;      float* Kb  = Qb + MN;
  float* Vb  = Kb + MN;      float* MSG = Vb + MN;
  float* MM  = MSG + MN;     float* D0  = MM + MN;
  float* D1  = D0 + MN;      float* H1  = D1 + MN;        // 2*MN
  float* ND  = H1 + 2 * MN;
  float* OD  = ND + (size_t)kB * kN * kN;
  float* Zb  = OD + (size_t)kB * kN * kN;

  auto gemm = [&](const float* A1, const float* A2, const float* W,
                  const float* bias, float* C, int K, int N, int relu) {
    dim3 grid(kM / 128, N / 64);
    mh_gemm_wmma_f32<<<grid, 256, 0, stream>>>(A1, A2, W, bias, C, K, N, relu);
  };
  auto lnorm = [&](const float* src, float* dst, const float* g, const float* b) {
    mh_layernorm<<<kM / 8, 256, 0, stream>>>(src, dst, g, b);
  };

  // 1) geometry distance maps
  mh_geom<<<kB, 64, 0, stream>>>(cam, params1, params2, ND, OD);

  // 2) input projections
  gemm(planeApp1, nullptr, app_W, app_b, X0, kD, kD, 0);
  gemm(planeApp2, nullptr, app_W, app_b, X1, kD, kD, 0);

  // 3) 18 GNN layers (even=self, odd=cross)
  for (int l = 0; l < kL; ++l) {
    const float* Wq = gWq + (size_t)l * kD * kD;
    const float* Wk = gWk + (size_t)l * kD * kD;
    const float* Wv = gWv + (size_t)l * kD * kD;
    const float* Wm = gWm + (size_t)l * kD * kD;
    const float* W1 = gW1 + (size_t)l * (2 * kD) * (2 * kD);
    const float* W2 = gW2 + (size_t)l * (2 * kD) * kD;
    const float* l1g = g1g + (size_t)l * kD; const float* l1b = g1b + (size_t)l * kD;
    const float* l2g = g2g + (size_t)l * kD; const float* l2b = g2b + (size_t)l * kD;
    const bool self = ((l & 1) == 0);
    for (int side = 0; side < 2; ++side) {
      const float* x   = side ? X1 : X0;
      const float* src = self ? x : (side ? X0 : X1);
      float* delta = side ? D1 : D0;
      gemm(x,   nullptr, Wq, nullptr, Qb, kD, kD, 0);
      gemm(src, nullptr, Wk, nullptr, Kb, kD, kD, 0);
      gemm(src, nullptr, Wv, nullptr, Vb, kD, kD, 0);
      mh_linattn<<<kB * kH, 256, 0, stream>>>(Qb, Kb, Vb, MSG);
      gemm(MSG, nullptr, Wm, nullptr, MM, kD, kD, 0);
      lnorm(MM, MM, l1g, l1b);
      gemm(x, MM, W1, nullptr, H1, kD, 2 * kD, 1);        // relu(cat @ W1), concat-free
      gemm(H1, nullptr, W2, nullptr, delta, 2 * kD, kD, 0);
      lnorm(delta, delta, l2g, l2b);
    }
    mh_add_inplace<<<(int)(MN / 256), 256, 0, stream>>>(X0, D0, (int)MN);
    mh_add_inplace<<<(int)(MN / 256), 256, 0, stream>>>(X1, D1, (int)MN);
  }

  // 4) descriptors (reuse Qb/Kb), scores + Z assembly
  gemm(X0, nullptr, desc_W, desc_b, Qb, kD, kD, 0);
  gemm(X1, nullptr, desc_W, desc_b, Kb, kD, kD, 0);
  const int ztot = kB * kZd * kZd;
  mh_scores<<<(ztot + 255) / 256, 256, 0, stream>>>(Qb, Kb, ND, OD, bin_score, Zb);

  // 5) Sinkhorn (LDS-resident) writes final output
  mh_sinkhorn<<<kB, 256, 0, stream>>>(Zb, (float*)d_out);
}